// AdvancedCS2GNN_77335181131973
// MI455X (gfx1250) — compile-verified
//
#include <hip/hip_runtime.h>
#include <math.h>

typedef __attribute__((ext_vector_type(2))) float v2f;
typedef __attribute__((ext_vector_type(8))) float v8f;

#define DIM 64
#define BN_EPS 1e-5f

// f32 WMMA: D(16x16,f32) = A(16x4,f32) x B(4x16,f32) + C
// A per lane: m = lane&15, k = (lane>>4)*2 + v  -> 2 VGPRs (v2f)
// B per lane: n = lane&15, k = (lane>>4)*2 + v  -> 2 VGPRs (v2f)
// C/D per lane: n = lane&15, m = v + 8*(lane>>4) -> 8 VGPRs (v8f)
__device__ __forceinline__ v8f wmma_f32_k4(v2f a, v2f b, v8f c) {
  return __builtin_amdgcn_wmma_f32_16x16x4_f32(
      /*neg_a=*/false, a, /*neg_b=*/false, b,
      /*c_mod=*/(short)0, c, /*reuse_a=*/false, /*reuse_b=*/false);
}

// ---------------- degree / norm ----------------
__global__ void k_init_deg(float* deg, int N) {
  int i = blockIdx.x * blockDim.x + threadIdx.x;
  if (i < N) deg[i] = 1.0f;  // self loop contributes 1
}

__global__ void k_edge_deg(float* deg, const int* __restrict__ dst, int E) {
  int i = blockIdx.x * blockDim.x + threadIdx.x;
  if (i < E) unsafeAtomicAdd(&deg[dst[i]], 1.0f);
}

__global__ void k_rsqrt_inplace(float* dis, int N) {
  int i = blockIdx.x * blockDim.x + threadIdx.x;
  if (i < N) dis[i] = rsqrtf(dis[i]);
}

__global__ void k_edge_norm(float* __restrict__ norm, const float* __restrict__ dis,
                            const int* __restrict__ src, const int* __restrict__ dst, int E) {
  int i = blockIdx.x * blockDim.x + threadIdx.x;
  if (i < E) norm[i] = dis[src[i]] * dis[dst[i]];
}

// ---------------- WMMA GEMM: Y = X(N x 64) @ W(64 x 64) [+ epilogue] ----------------
// MODE 0: Y = acc            (feeds scatter)
// MODE 1: Y = tanh(acc + b)  (predictor layer 1)
// MODE 2: Y = acc + b        (final output)
template <int MODE>
__global__ __launch_bounds__(256) void k_gemm64(const float* __restrict__ X,
                                                const float* __restrict__ W,
                                                const float* __restrict__ bias,
                                                float* __restrict__ Y,
                                                int N, int nTiles) {
  // Pair-interleaved weight staging: Wl[k2*64 + j] = { W[2*k2][j], W[2*k2+1][j] }
  // -> each WMMA B operand is one aligned ds_load_b64 (no register re-packing).
  __shared__ v2f Wl[(DIM / 2) * DIM];
  for (int i = threadIdx.x; i < (DIM / 2) * DIM; i += 256) {
    int k2 = i >> 6;
    int j  = i & 63;
    v2f w;
    w.x = W[(2 * k2) * DIM + j];
    w.y = W[(2 * k2 + 1) * DIM + j];
    Wl[i] = w;
  }
  __syncthreads();

  const int wave = threadIdx.x >> 5;
  const int lane = threadIdx.x & 31;
  const int tile = blockIdx.x * 8 + wave;
  if (tile >= nTiles) return;  // wave-uniform: EXEC stays all-ones for WMMA

  const int m  = lane & 15;    // A row within tile / B,C column
  const int hi = lane >> 4;    // 0 or 1

  int arow = tile * 16 + m;
  if (arow >= N) arow = N - 1;  // clamp (guarded on store)
  const float* xr = X + (size_t)arow * DIM;

  v8f acc0 = {}, acc1 = {}, acc2 = {}, acc3 = {};
#pragma unroll
  for (int kk = 0; kk < 16; ++kk) {
    const int k0 = (kk << 2) + (hi << 1);  // even
    v2f a;
    a.x = xr[k0];
    a.y = xr[k0 + 1];
    const v2f* wrow = &Wl[(size_t)(k0 >> 1) * DIM];  // (2*kk + hi) * 64
    acc0 = wmma_f32_k4(a, wrow[m],      acc0);
    acc1 = wmma_f32_k4(a, wrow[m + 16], acc1);
    acc2 = wmma_f32_k4(a, wrow[m + 32], acc2);
    acc3 = wmma_f32_k4(a, wrow[m + 48], acc3);
  }

  const int rbase = tile * 16 + hi * 8;
#pragma unroll
  for (int v = 0; v < 8; ++v) {
    const int row = rbase + v;
    if (row >= N) continue;
    float* yr = Y + (size_t)row * DIM;
    float v0 = acc0[v], v1 = acc1[v], v2 = acc2[v], v3 = acc3[v];
    if (MODE == 1) {
      v0 = tanhf(v0 + bias[m]);
      v1 = tanhf(v1 + bias[m + 16]);
      v2 = tanhf(v2 + bias[m + 32]);
      v3 = tanhf(v3 + bias[m + 48]);
    } else if (MODE == 2) {
      v0 += bias[m];
      v1 += bias[m + 16];
      v2 += bias[m + 32];
      v3 += bias[m + 48];
    }
    yr[m]      = v0;
    yr[m + 16] = v1;
    yr[m + 32] = v2;
    yr[m + 48] = v3;
  }
}

// ---------------- aggregation ----------------
// Seed AGG with the self-loop term (also serves as the zero-init each layer).
__global__ void k_selfloop(float* __restrict__ agg, const float* __restrict__ H,
                           const float* __restrict__ dis, int total) {
  int t = blockIdx.x * blockDim.x + threadIdx.x;
  if (t < total) {
    float s = dis[t >> 6];
    agg[t] = H[t] * (s * s);
  }
}

// One thread per (edge, channel pair): agg[dst, c:c+2] += H[src, c:c+2] * norm[e]
__global__ void k_scatter(float* __restrict__ agg, const float* __restrict__ H,
                          const int* __restrict__ src, const int* __restrict__ dst,
                          const float* __restrict__ norm, int E) {
  int t = blockIdx.x * blockDim.x + threadIdx.x;
  int e = t >> 5;
  if (e < E) {
    int c  = (t & 31) << 1;
    float nr = norm[e];
    const float2* hp = (const float2*)(H + (size_t)src[e] * DIM + c);
    float2 hv = *hp;
    float* ap = agg + (size_t)dst[e] * DIM + c;
    unsafeAtomicAdd(ap,     hv.x * nr);
    unsafeAtomicAdd(ap + 1, hv.y * nr);
  }
}

// In-place: agg = relu( (agg + bias - m) * g * rsqrt(v+eps) + b )
__global__ void k_bias_bn_relu(float* __restrict__ agg, const float* __restrict__ bias,
                               const float* __restrict__ g, const float* __restrict__ b,
                               const float* __restrict__ m, const float* __restrict__ v,
                               int total) {
  int t = blockIdx.x * blockDim.x + threadIdx.x;
  if (t < total) {
    int c = t & 63;
    float s   = g[c] * rsqrtf(v[c] + BN_EPS);
    float val = (agg[t] + bias[c] - m[c]) * s + b[c];
    agg[t] = val > 0.0f ? val : 0.0f;
  }
}

__global__ void k_bias(float* __restrict__ agg, const float* __restrict__ bias, int total) {
  int t = blockIdx.x * blockDim.x + threadIdx.x;
  if (t < total) agg[t] += bias[t & 63];
}

// ---------------- host launcher ----------------
extern "C" void kernel_launch(void* const* d_in, const int* in_sizes, int n_in,
                              void* d_out, int out_size, void* d_ws, size_t ws_size,
                              hipStream_t stream) {
  const float* x   = (const float*)d_in[0];
  const int*   ei  = (const int*)d_in[1];   // edge_index [2, E]
  const float* W1  = (const float*)d_in[2];
  const float* b1  = (const float*)d_in[3];
  const float* W2  = (const float*)d_in[4];
  const float* b2  = (const float*)d_in[5];
  const float* W3  = (const float*)d_in[6];
  const float* b3  = (const float*)d_in[7];
  const float* bn1g = (const float*)d_in[8];
  const float* bn1b = (const float*)d_in[9];
  const float* bn1m = (const float*)d_in[10];
  const float* bn1v = (const float*)d_in[11];
  const float* bn2g = (const float*)d_in[12];
  const float* bn2b = (const float*)d_in[13];
  const float* bn2m = (const float*)d_in[14];
  const float* bn2v = (const float*)d_in[15];
  const float* pW1 = (const float*)d_in[16];
  const float* pb1 = (const float*)d_in[17];
  const float* pW2 = (const float*)d_in[18];
  const float* pb2 = (const float*)d_in[19];

  const int N = in_sizes[0] / DIM;
  const int E = in_sizes[1] / 2;
  const int* src = ei;
  const int* dst = ei + E;

  // workspace layout (floats): dis[N] | norm[E] | H[N*64] | AGG[N*64]
  float* ws   = (float*)d_ws;
  float* dis  = ws;
  float* norm = dis + N;
  float* H    = norm + E;
  float* AGG  = H + (size_t)N * DIM;

  const int total = N * DIM;
  const int TB = 256;
  const int nTiles   = (N + 15) / 16;
  const int gemmBlks = (nTiles + 7) / 8;
  const int nBlk     = (N + TB - 1) / TB;
  const int eBlk     = (E + TB - 1) / TB;
  const int ndBlk    = (total + TB - 1) / TB;
  const int scBlk    = (int)(((size_t)E * 32 + TB - 1) / TB);

  // degrees + symmetric norm
  k_init_deg<<<nBlk, TB, 0, stream>>>(dis, N);
  k_edge_deg<<<eBlk, TB, 0, stream>>>(dis, dst, E);
  k_rsqrt_inplace<<<nBlk, TB, 0, stream>>>(dis, N);
  k_edge_norm<<<eBlk, TB, 0, stream>>>(norm, dis, src, dst, E);

  // GCN layer 1
  k_gemm64<0><<<gemmBlks, TB, 0, stream>>>(x, W1, nullptr, H, N, nTiles);
  k_selfloop<<<ndBlk, TB, 0, stream>>>(AGG, H, dis, total);
  k_scatter<<<scBlk, TB, 0, stream>>>(AGG, H, src, dst, norm, E);
  k_bias_bn_relu<<<ndBlk, TB, 0, stream>>>(AGG, b1, bn1g, bn1b, bn1m, bn1v, total);

  // GCN layer 2
  k_gemm64<0><<<gemmBlks, TB, 0, stream>>>(AGG, W2, nullptr, H, N, nTiles);
  k_selfloop<<<ndBlk, TB, 0, stream>>>(AGG, H, dis, total);
  k_scatter<<<scBlk, TB, 0, stream>>>(AGG, H, src, dst, norm, E);
  k_bias_bn_relu<<<ndBlk, TB, 0, stream>>>(AGG, b2, bn2g, bn2b, bn2m, bn2v, total);

  // GCN layer 3 (bias only, no BN/ReLU)
  k_gemm64<0><<<gemmBlks, TB, 0, stream>>>(AGG, W3, nullptr, H, N, nTiles);
  k_selfloop<<<ndBlk, TB, 0, stream>>>(AGG, H, dis, total);
  k_scatter<<<scBlk, TB, 0, stream>>>(AGG, H, src, dst, norm, E);
  k_bias<<<ndBlk, TB, 0, stream>>>(AGG, b3, total);

  // predictor MLP: tanh(AGG @ pW1 + pb1) @ pW2 + pb2
  k_gemm64<1><<<gemmBlks, TB, 0, stream>>>(AGG, pW1, pb1, H, N, nTiles);
  k_gemm64<2><<<gemmBlks, TB, 0, stream>>>(H, pW2, pb2, (float*)d_out, N, nTiles);
}